// MovingAverage_1580547966243
// MI455X (gfx1250) — compile-verified
//
#include <hip/hip_runtime.h>
#include <cstdint>

// Problem constants (fixed by the reference).
#define B_N 64
#define T_N 8192
#define D_N 128
#define LW 12
#define RW 12
#define WIN 24

// Blocking: one block = 1 batch row x TC time steps, 256 threads = 8 waves.
#define TC 128
#define ROWS (TC + WIN)     // 152 staged rows (12-halo each side)
#define LSTRIDE 136         // padded LDS row stride in floats (bank-conflict-free)

typedef __attribute__((ext_vector_type(2))) float v2f;
typedef __attribute__((ext_vector_type(4))) float v4f;
typedef __attribute__((ext_vector_type(8))) float v8f;

// Exact pointee type the async builtin expects: int __vector(4) (vector_size).
typedef int i4 __attribute__((vector_size(16)));
typedef __attribute__((address_space(1))) i4* as1_i4p;
typedef __attribute__((address_space(3))) i4* as3_i4p;

#if __has_builtin(__builtin_amdgcn_wmma_f32_16x16x4_f32)
#define USE_WMMA 1
#else
#define USE_WMMA 0
#endif

__device__ __forceinline__ void async_copy_b128(float* ldsDst, const float* gSrc) {
#if __has_builtin(__builtin_amdgcn_global_load_async_to_lds_b128)
  __builtin_amdgcn_global_load_async_to_lds_b128(
      (as1_i4p)(uintptr_t)gSrc,
      (as3_i4p)(unsigned int)(uintptr_t)ldsDst,
      0, 0);
#else
  asm volatile("global_load_async_to_lds_b128 %0, %1, off"
               :
               : "v"((unsigned int)(uintptr_t)ldsDst),
                 "v"((unsigned long long)(uintptr_t)gSrc)
               : "memory");
#endif
}

__device__ __forceinline__ void wait_async_zero() {
#if __has_builtin(__builtin_amdgcn_s_wait_asynccnt)
  __builtin_amdgcn_s_wait_asynccnt(0);
#else
  asm volatile("s_wait_asynccnt 0" ::: "memory");
#endif
}

__global__ __launch_bounds__(256)
void moving_avg_kernel(const float* __restrict__ data, float* __restrict__ out) {
  __shared__ float lds[ROWS * LSTRIDE];

  const int tid = threadIdx.x;
  const int blk = blockIdx.x;
  const int b   = blk / (T_N / TC);
  const int t0  = (blk - b * (T_N / TC)) * TC;

  const float* src = data + (size_t)b * T_N * D_N;

  // ---- Stage [t0-12, t0+TC+12) x D into LDS via async B128 copies.
  // Out-of-range halo rows are zero-filled so clamped windows fall out for free.
  for (int idx = tid; idx < ROWS * (D_N / 4); idx += 256) {
    const int row = idx >> 5;          // D_N/4 == 32 float4 per row
    const int c4  = (idx & 31) << 2;   // column in floats
    const int gt  = t0 - LW + row;
    float* dst = &lds[row * LSTRIDE + c4];
    if ((unsigned)gt < (unsigned)T_N) {
      async_copy_b128(dst, src + (size_t)gt * D_N + c4);
    } else {
      v4f z = {};
      *(v4f*)dst = z;
    }
  }
  wait_async_zero();
  __syncthreads();

  // ---- Per-wave: 16 channels, all 8 output t-tiles of this chunk.
  const int wave  = tid >> 5;
  const int lane  = tid & 31;
  const int n     = lane & 15;         // A row (M) and C/D column (N)
  const int khalf = lane >> 4;
  const int col   = (wave << 4) + n;   // channel index d == LDS column
  const int kk0   = khalf << 1;        // A/B K sub-index held in v0 (v1 = kk0+1)

#if USE_WMMA
  // Band-matrix A fragments: position-independent, hoisted out of the tile loop.
  v2f afr[10];
#pragma unroll
  for (int j = 0; j < 10; ++j) {
    afr[j].x = ((unsigned)(4 * j + kk0     - n) < (unsigned)WIN) ? 1.0f : 0.0f;
    afr[j].y = ((unsigned)(4 * j + kk0 + 1 - n) < (unsigned)WIN) ? 1.0f : 0.0f;
  }
#endif

  // Uniform per-block output bases (SGPR saddr) + 32-bit element offsets.
  float* outS = out + ((size_t)b * T_N + t0) * D_N;
  float* outR = outS + (size_t)B_N * T_N * D_N;
  const bool interior = (t0 >= LW) && (t0 + TC + RW <= T_N);  // uniform branch

  for (int tt = 0; tt < TC / 16; ++tt) {
    const int kbase = tt << 4;         // LDS row of k-range start (== t_tile-12)
    v8f acc = {};

#if USE_WMMA
    // sums(16x16) = sum_j A_j(16x4 band) * X_j(4x16), 10 steps cover k in [0,40)
#pragma unroll
    for (int j = 0; j < 10; ++j) {
      v2f bf;
      const int kr = kbase + 4 * j + kk0;
      bf.x = lds[kr * LSTRIDE + col];
      bf.y = lds[(kr + 1) * LSTRIDE + col];
      acc = __builtin_amdgcn_wmma_f32_16x16x4_f32(
          false, afr[j], false, bf, (short)0, acc, false, false);
    }
#else
#pragma unroll
    for (int r = 0; r < 8; ++r) {
      const int mloc = r + (khalf << 3);
      float s = 0.0f;
      for (int k = 0; k < WIN; ++k)
        s += lds[(kbase + mloc + k) * LSTRIDE + col];
      acc[r] = s;
    }
#endif

    // C/D layout: VGPR r holds (M = r + 8*khalf, N = lane&15).
    const int mbase   = khalf << 3;
    const int rowbase = kbase + LW + mbase;          // LDS row of output r=0
    const int offbase = (kbase + mbase) * D_N + col; // element offset, < 2^15

    if (interior) {
#pragma unroll
      for (int r = 0; r < 8; ++r) {
        const float x = lds[(rowbase + r) * LSTRIDE + col];
        const float s = acc[r] * (1.0f / (float)WIN);
        __builtin_nontemporal_store(s,     outS + offbase + r * D_N);
        __builtin_nontemporal_store(x - s, outR + offbase + r * D_N);
      }
    } else {
#pragma unroll
      for (int r = 0; r < 8; ++r) {
        const int t     = t0 + kbase + mbase + r;
        const float x   = lds[(rowbase + r) * LSTRIDE + col];
        const int left  = (t - LW) < 0 ? 0 : (t - LW);
        const int right = (t + RW) > T_N ? T_N : (t + RW);
        const float s   = acc[r] / (float)(right - left);
        __builtin_nontemporal_store(s,     outS + offbase + r * D_N);
        __builtin_nontemporal_store(x - s, outR + offbase + r * D_N);
      }
    }
  }
}

extern "C" void kernel_launch(void* const* d_in, const int* in_sizes, int n_in,
                              void* d_out, int out_size, void* d_ws, size_t ws_size,
                              hipStream_t stream) {
  (void)in_sizes; (void)n_in; (void)out_size; (void)d_ws; (void)ws_size;
  const float* data = (const float*)d_in[0];
  float* out = (float*)d_out;
  dim3 grid(B_N * (T_N / TC));
  dim3 block(256);
  moving_avg_kernel<<<grid, block, 0, stream>>>(data, out);
}